// MultiHeadAttention_8761733284075
// MI455X (gfx1250) — compile-verified
//
#include <hip/hip_runtime.h>
#include <math.h>

// ---------------------------------------------------------------------------
// MI455X / gfx1250 fused multi-head attention forward.
// Bandwidth-bound op (attn output = 256MB) -> keep fp32 numerics and use
// V_WMMA_F32_16X16X4_F32 for every GEMM stage.
// ---------------------------------------------------------------------------

typedef __attribute__((ext_vector_type(2))) float v2f;
typedef __attribute__((ext_vector_type(8))) float v8f;

// Fixed problem sizes from the reference.
#define BB   4
#define LL   1024
#define DD   1024
#define HH   16
#define DKV  64          // DK == DV == 64
#define MM   (BB * LL)   // 4096 rows

__device__ __forceinline__ v8f wmma4(v2f a, v2f b, v8f c) {
  // D = A(16x4 f32) * B(4x16 f32) + C(16x16 f32)
  return __builtin_amdgcn_wmma_f32_16x16x4_f32(
      /*neg_a=*/false, a, /*neg_b=*/false, b,
      /*c_mod=*/(short)0, c, /*reuse_a=*/false, /*reuse_b=*/false);
}

// ---------------------------------------------------------------------------
// 1) Per-row LayerNorm statistics (mean, rstd) for q,k,v. One wave per row.
// ---------------------------------------------------------------------------
__global__ void ln_stats_kernel(const float* __restrict__ q,
                                const float* __restrict__ k,
                                const float* __restrict__ v,
                                float* __restrict__ stats) {
  int tid  = blockIdx.x * blockDim.x + threadIdx.x;
  int wave = tid >> 5, lane = tid & 31;
  if (wave >= 3 * MM) return;
  int tensor = wave / MM, r = wave - tensor * MM;
  const float* src = (tensor == 0) ? q : ((tensor == 1) ? k : v);
  const float* row = src + (size_t)r * DD;
  float s = 0.f, s2 = 0.f;
  for (int i = lane; i < DD; i += 32) { float x = row[i]; s += x; s2 += x * x; }
  for (int off = 16; off > 0; off >>= 1) {
    s  += __shfl_xor(s,  off, 32);
    s2 += __shfl_xor(s2, off, 32);
  }
  if (lane == 0) {
    float mean = s * (1.f / DD);
    float var  = s2 * (1.f / DD) - mean * mean;
    stats[(size_t)(tensor * MM + r) * 2 + 0] = mean;
    stats[(size_t)(tensor * MM + r) * 2 + 1] = rsqrtf(var + 1e-5f);
  }
}

// ---------------------------------------------------------------------------
// 2/5) out[MxN] = LN(A)[MxK] @ W[NxK]^T.  USE_LN is a compile-time flag so
// the fc1 instantiation carries no LN code and the projection instantiation
// has a branch-free inner loop (LN math fuses into the load stream).
// One wave computes a 16(M) x 64(N) tile: 4 independent accumulator chains,
// A-frag reused x4.  Both A and W are row-major over K, so A- and B-fragments
// load identically: lane l -> float2 at [tile_row = l%16][k + 2*(l/16)].
// ---------------------------------------------------------------------------
template <bool USE_LN>
__global__ void gemm_abt_ln_kernel(const float* __restrict__ A,
                                   const float* __restrict__ W,
                                   const float* __restrict__ stats,
                                   const float* __restrict__ lnw,
                                   const float* __restrict__ lnb,
                                   float* __restrict__ out,
                                   int M, int N, int K) {
  int tid  = blockIdx.x * blockDim.x + threadIdx.x;
  int wave = tid >> 5, lane = tid & 31;
  int nt64 = N >> 6;
  int mt = wave / nt64, nt = wave - mt * nt64;
  if (mt >= (M >> 4)) return;
  int m0 = mt << 4, n0 = nt << 6;
  int row = lane & 15, khalf = lane >> 4;

  float mean = 0.f, rstd = 1.f;
  if (USE_LN) {
    mean = stats[(size_t)(m0 + row) * 2 + 0];
    rstd = stats[(size_t)(m0 + row) * 2 + 1];
  }
  const float* arow = A + (size_t)(m0 + row) * K;
  const float* wrow = W + (size_t)(n0 + row) * K;

  v8f acc0 = {}, acc1 = {}, acc2 = {}, acc3 = {};
  #pragma unroll 2
  for (int kk = 0; kk < K; kk += 4) {
    int kb = kk + (khalf << 1);
    v2f a = *(const v2f*)(arow + kb);
    if (USE_LN) {
      a.x = (a.x - mean) * rstd * lnw[kb]     + lnb[kb];
      a.y = (a.y - mean) * rstd * lnw[kb + 1] + lnb[kb + 1];
    }
    v2f b0 = *(const v2f*)(wrow + kb);
    v2f b1 = *(const v2f*)(wrow + (size_t)16 * K + kb);
    v2f b2 = *(const v2f*)(wrow + (size_t)32 * K + kb);
    v2f b3 = *(const v2f*)(wrow + (size_t)48 * K + kb);
    acc0 = wmma4(a, b0, acc0);
    acc1 = wmma4(a, b1, acc1);
    acc2 = wmma4(a, b2, acc2);
    acc3 = wmma4(a, b3, acc3);
  }
  #pragma unroll
  for (int r = 0; r < 8; ++r) {
    float* orow = out + (size_t)(m0 + r + (khalf << 3)) * N + n0 + row;
    orow[0]  = acc0[r];
    orow[16] = acc1[r];
    orow[32] = acc2[r];
    orow[48] = acc3[r];
  }
}

// ---------------------------------------------------------------------------
// 3) S = Q K^T / 8, mask (-1e32), softmax, post-softmax zeroing, store p.
// Block = 128 threads (4 waves) owns 16 q-rows of one (h,b); the full
// 16 x 1024 logit strip lives in 64KB LDS.
// Occupancy is LDS-capped (5 blocks / 320KB WGP), NOT register-capped, so
// __launch_bounds__(128, 1) relaxes the VGPR budget and lets the compiler
// keep all 16 batched K-fragments live (clause + staged waits) instead of
// recycling one register pair with a full wait per WMMA.
// ---------------------------------------------------------------------------
__global__ void __launch_bounds__(128, 1)
attn_softmax_kernel(const float* __restrict__ qh,
                    const float* __restrict__ kh,
                    const int* __restrict__ mask,
                    float* __restrict__ attn) {
  __shared__ float sblk[16 * LL];  // 64 KB
  int qt = blockIdx.x;             // q tile: 0..63
  int hb = blockIdx.y;             // h*B + b: 0..63
  int h = hb >> 2, b = hb & 3;
  int t = threadIdx.x;
  int wave = t >> 5, lane = t & 31;
  int row = lane & 15, khalf = lane >> 4;
  int q0 = qt << 4;

  // Preload Q fragments for the full K=64 inner dim (16 steps of 4).
  const float* qrow = qh + (size_t)(b * LL + q0 + row) * DD + h * DKV;
  v2f qf[16];
  #pragma unroll
  for (int i = 0; i < 16; ++i)
    qf[i] = *(const v2f*)(qrow + i * 4 + (khalf << 1));

  // Each wave computes 16 of the 64 k-tiles (strided by 4).
  #pragma unroll 2
  for (int i = 0; i < 16; ++i) {
    int k0 = (wave + (i << 2)) << 4;
    const float* krow = kh + (size_t)(b * LL + k0 + row) * DD + h * DKV;

    // Batched fragment loads (independent -> single clause + staged waits).
    v2f kf[16];
    #pragma unroll
    for (int s = 0; s < 16; ++s)
      kf[s] = *(const v2f*)(krow + s * 4 + (khalf << 1));

    // Two independent WMMA chains halve the RAW dependency depth.
    v8f accE = {}, accO = {};
    #pragma unroll
    for (int s = 0; s < 16; s += 2) {
      accE = wmma4(qf[s],     kf[s],     accE);
      accO = wmma4(qf[s + 1], kf[s + 1], accO);
    }
    #pragma unroll
    for (int r = 0; r < 8; ++r)
      sblk[(r + (khalf << 3)) * LL + k0 + row] =
          (accE[r] + accO[r]) * 0.125f;                 // /sqrt(64)
  }
  __syncthreads();

  // Softmax: 8 threads per row (contiguous lanes -> width-8 shuffles).
  int srow = t >> 3, g = t & 7;
  const int* mrow = mask + (size_t)b * LL * LL + (size_t)(q0 + srow) * LL;
  float* sr = sblk + srow * LL;
  float lmax = -3.402823466e38f;
  for (int kx = g; kx < LL; kx += 8) {
    float ml = (mrow[kx] != 0) ? -1e32f : sr[kx];
    sr[kx] = ml;
    lmax = fmaxf(lmax, ml);
  }
  for (int off = 1; off < 8; off <<= 1)
    lmax = fmaxf(lmax, __shfl_xor(lmax, off, 8));
  float lsum = 0.f;
  for (int kx = g; kx < LL; kx += 8) {
    float e = __expf(sr[kx] - lmax);  // all-masked row: exp(0)=1, matches ref
    sr[kx] = e;
    lsum += e;
  }
  for (int off = 1; off < 8; off <<= 1)
    lsum += __shfl_xor(lsum, off, 8);
  float inv = 1.f / lsum;
  for (int kx = g; kx < LL; kx += 8) {
    float p = sr[kx] * inv;
    if (mrow[kx] != 0) p = 0.f;     // p = p * (~m)
    sr[kx] = p;
  }
  __syncthreads();

  // Coalesced 128-wide store of the probability strip.
  float* arow = attn + (size_t)hb * LL * LL + (size_t)q0 * LL;
  for (int rr = 0; rr < 16; ++rr)
    for (int kx = t; kx < LL; kx += 128)
      arow[(size_t)rr * LL + kx] = sblk[rr * LL + kx];
}

// ---------------------------------------------------------------------------
// 4) ctx[b,q,h,:] = P[hb] @ V_h.  One wave: 16(q) x 64(dv) tile, K-loop 1024,
// 4 independent accumulator chains.  (No launch_bounds: this kernel streams
// 256MB from HBM and needs maximum wave occupancy for latency hiding.)
// ---------------------------------------------------------------------------
__global__ void pv_kernel(const float* __restrict__ attn,
                          const float* __restrict__ vh,
                          float* __restrict__ ctx) {
  int tid  = blockIdx.x * blockDim.x + threadIdx.x;
  int wave = tid >> 5, lane = tid & 31;
  int hb = wave >> 6;          // 0..63
  int qt = wave & 63;          // 0..63
  if (hb >= HH * BB) return;
  int h = hb >> 2, b = hb & 3;
  int q0 = qt << 4;
  int row = lane & 15, khalf = lane >> 4;

  const float* prow  = attn + (size_t)hb * LL * LL + (size_t)(q0 + row) * LL;
  const float* vbase = vh + (size_t)(b * LL) * DD + h * DKV + row; // N = row

  v8f acc0 = {}, acc1 = {}, acc2 = {}, acc3 = {};
  #pragma unroll 2
  for (int kk = 0; kk < LL; kk += 4) {
    int kb = kk + (khalf << 1);
    v2f a = *(const v2f*)(prow + kb);
    const float* v0 = vbase + (size_t)kb * DD;       // V row kb
    const float* v1 = v0 + DD;                       // V row kb+1
    v2f b0 = { v0[0],  v1[0]  };
    v2f b1 = { v0[16], v1[16] };
    v2f b2 = { v0[32], v1[32] };
    v2f b3 = { v0[48], v1[48] };
    acc0 = wmma4(a, b0, acc0);
    acc1 = wmma4(a, b1, acc1);
    acc2 = wmma4(a, b2, acc2);
    acc3 = wmma4(a, b3, acc3);
  }
  #pragma unroll
  for (int r = 0; r < 8; ++r) {
    float* orow = ctx + (size_t)(b * LL + q0 + r + (khalf << 3)) * DD
                      + h * DKV + row;
    orow[0]  = acc0[r];
    orow[16] = acc1[r];
    orow[32] = acc2[r];
    orow[48] = acc3[r];
  }
}

// ---------------------------------------------------------------------------
extern "C" void kernel_launch(void* const* d_in, const int* in_sizes, int n_in,
                              void* d_out, int out_size, void* d_ws, size_t ws_size,
                              hipStream_t stream) {
  (void)in_sizes; (void)n_in; (void)out_size; (void)ws_size;
  const float* q    = (const float*)d_in[0];
  const float* k    = (const float*)d_in[1];
  const float* v    = (const float*)d_in[2];
  // d_in[3] = diag_mask (unused by reference math)
  const int*   mask = (const int*)  d_in[4];
  const float* ln1w = (const float*)d_in[5];
  const float* ln1b = (const float*)d_in[6];
  const float* ln2w = (const float*)d_in[7];
  const float* ln2b = (const float*)d_in[8];
  const float* ln3w = (const float*)d_in[9];
  const float* ln3b = (const float*)d_in[10];
  const float* wq   = (const float*)d_in[11];
  const float* wk   = (const float*)d_in[12];
  const float* wv   = (const float*)d_in[13];
  const float* wfc  = (const float*)d_in[14];

  // Workspace layout (floats): stats[3*4096*2] | qh | kh | vh | ctx  (~67 MB)
  float* ws    = (float*)d_ws;
  float* stats = ws;
  float* qh    = ws + 3 * MM * 2;
  float* kh    = qh + (size_t)MM * DD;
  float* vh    = kh + (size_t)MM * DD;
  float* ctx   = vh + (size_t)MM * DD;

  float* dynamic = (float*)d_out;                      // [B, L, D]
  float* attn    = dynamic + (size_t)MM * DD;          // [H*B, L, L]

  // 1) LN stats: 3*4096 waves; 256 threads = 8 waves/block.
  ln_stats_kernel<<<(3 * MM) / 8, 256, 0, stream>>>(q, k, v, stats);

  // 2) Projections: (4096/16)*(1024/64) = 4096 waves -> 1024 blocks of 128.
  gemm_abt_ln_kernel<true><<<1024, 128, 0, stream>>>(q, wq, stats + 0 * MM * 2,
                                                     ln1w, ln1b, qh, MM, DD, DD);
  gemm_abt_ln_kernel<true><<<1024, 128, 0, stream>>>(k, wk, stats + 1 * MM * 2,
                                                     ln2w, ln2b, kh, MM, DD, DD);
  gemm_abt_ln_kernel<true><<<1024, 128, 0, stream>>>(v, wv, stats + 2 * MM * 2,
                                                     ln3w, ln3b, vh, MM, DD, DD);

  // 3) QK^T + mask + softmax -> attn (64 q-tiles x 64 (h,b) pairs).
  dim3 g3(LL / 16, HH * BB);
  attn_softmax_kernel<<<g3, 128, 0, stream>>>(qh, kh, mask, attn);

  // 4) P @ V -> ctx: 64*64 = 4096 waves -> 1024 blocks of 128.
  pv_kernel<<<1024, 128, 0, stream>>>(attn, vh, ctx);

  // 5) fc1: dynamic = ctx @ wfc^T (no LN).
  gemm_abt_ln_kernel<false><<<1024, 128, 0, stream>>>(ctx, wfc, nullptr,
                                                      nullptr, nullptr, dynamic,
                                                      MM, DD, DD);
}